// SDF_51780125721345
// MI455X (gfx1250) — compile-verified
//
#include <hip/hip_runtime.h>

typedef __attribute__((ext_vector_type(2))) float v2f;
typedef __attribute__((ext_vector_type(8))) float v8f;

#define NPOS   3
#define VD     4
#define NLEV   24
#define CAP    (1 << 18)
#define INDIM  51
#define KPAD   52
#define HID    32
#define OUTD   33
#define GEOM   32

__device__ __forceinline__ float gelu_exact(float x) {
    return 0.5f * x * (1.0f + erff(x * 0.70710678118654752f));
}

// One mid layer: Y(16x32) = gelu( X(16x32) @ W(32x32) + B ), via V_WMMA_F32_16X16X4_F32.
// lo = lane&15, hi = lane>>4. A operand: lane holds X[lo][k0+2*hi .. +1]; B operand:
// lane holds W[k0+2*hi][nb+lo], W[k0+2*hi+1][nb+lo].
__device__ __forceinline__ void mid_layer(const float (*__restrict__ X)[34],
                                          const float* __restrict__ W,
                                          const float* __restrict__ B,
                                          float (*__restrict__ Y)[34],
                                          int lo, int hi) {
    for (int nb = 0; nb < HID; nb += 16) {
        v8f acc = {0.f, 0.f, 0.f, 0.f, 0.f, 0.f, 0.f, 0.f};
        const int col = nb + lo;
#pragma unroll
        for (int k0 = 0; k0 < HID; k0 += 4) {
            const int kb = k0 + 2 * hi;
            v2f a = *(const v2f*)&X[lo][kb];           // 8B aligned: even stride, even kb
            v2f b;
            b.x = W[kb * HID + col];
            b.y = W[(kb + 1) * HID + col];
            acc = __builtin_amdgcn_wmma_f32_16x16x4_f32(false, a, false, b,
                                                        (short)0, acc, false, false);
        }
        const float bias = B[col];
#pragma unroll
        for (int r = 0; r < 8; ++r)
            Y[8 * hi + r][col] = gelu_exact(acc[r] + bias);
    }
}

__launch_bounds__(128, 1)
__global__ void sdf_permuto_fused(const float* __restrict__ points,
                                  const int*   __restrict__ iter_nr_p,
                                  const float* __restrict__ tables,
                                  const float* __restrict__ shifts,
                                  const float* __restrict__ w1, const float* __restrict__ b1,
                                  const float* __restrict__ w2, const float* __restrict__ b2,
                                  const float* __restrict__ w3, const float* __restrict__ b3,
                                  const float* __restrict__ w4, const float* __restrict__ b4,
                                  float* __restrict__ out, int N)
{
    __shared__ float feat[4][32][64];     // per-wave 32 rows x (51 feat + pad)
    __shared__ float act[4][2][16][34];   // per-wave ping-pong activations

    const int wv   = threadIdx.x >> 5;
    const int lane = threadIdx.x & 31;
    const int p    = blockIdx.x * 128 + threadIdx.x;
    const bool valid = p < N;

    // ---------------- Phase 1: permutohedral encoding (one point per lane) ----------
    const float px = valid ? points[3 * p + 0] : 0.f;
    const float py = valid ? points[3 * p + 1] : 0.f;
    const float pz = valid ? points[3 * p + 2] : 0.f;

    const int   it    = iter_nr_p[0];
    const float t     = fminf(fmaxf((float)it * 1e-4f, 0.f), 1.f);
    const float alpha = (0.3f + 0.7f * t) * (float)NLEV;

    // SF = INV_STD / sqrt((i+1)(i+2)), INV_STD = 4*sqrt(2/3)
    const float SF0 = 2.3094010767585034f;   // 4/sqrt(3)
    const float SF1 = 1.3333333333333333f;
    const float SF2 = 0.9428090415820634f;   // 2*sqrt(2)/3

    float* frow = feat[wv][lane];

    for (int l = 0; l < NLEV; ++l) {
        const float xw = fminf(fmaxf(alpha - (float)l, 0.f), 1.f);
        const float wl = 0.5f * (1.0f - cosf(3.14159265358979323f * xw));
        // scales = geomspace(1, 1e-4, 24)  ->  1/scale[l] = 10^(4l/23)
        const float inv_scale = exp2f((float)l * (4.0f / 23.0f) * 3.3219280948873623f);

        const float q0 = px * inv_scale + shifts[3 * l + 0];
        const float q1 = py * inv_scale + shifts[3 * l + 1];
        const float q2 = pz * inv_scale + shifts[3 * l + 2];

        const float cf0 = q0 * SF0, cf1 = q1 * SF1, cf2 = q2 * SF2;

        float e[VD];
        e[0] = cf0 + cf1 + cf2;
        e[1] = cf1 + cf2 - cf0;
        e[2] = cf2 - 2.f * cf1;
        e[3] = -3.f * cf2;

        float r0[VD], df[VD];
        float sum4 = 0.f;
#pragma unroll
        for (int i = 0; i < VD; ++i) { r0[i] = rintf(e[i] * 0.25f) * 4.f; sum4 += r0[i]; }
        const int sum_ = (int)rintf(sum4 * 0.25f);
#pragma unroll
        for (int i = 0; i < VD; ++i) df[i] = e[i] - r0[i];

        int rk[VD];
#pragma unroll
        for (int i = 0; i < VD; ++i) {
            int r = 0;
#pragma unroll
            for (int j = 0; j < VD; ++j)
                if (df[j] > df[i] || (df[j] == df[i] && j < i)) r++;
            rk[i] = r + sum_;
        }
#pragma unroll
        for (int i = 0; i < VD; ++i) {
            if (rk[i] < 0)         { rk[i] += VD; r0[i] += 4.f; }
            else if (rk[i] > NPOS) { rk[i] -= VD; r0[i] -= 4.f; }
        }

        float dl[VD];
#pragma unroll
        for (int i = 0; i < VD; ++i) dl[i] = (e[i] - r0[i]) * 0.25f;

        // barycentric: bary[3-rank] += delta; bary[4-rank] -= delta
        float bb0 = 0.f, bb1 = 0.f, bb2 = 0.f, bb3 = 0.f, bb4 = 0.f;
#pragma unroll
        for (int i = 0; i < VD; ++i) {
            const float d = dl[i];
            const int a = NPOS - rk[i];
            bb0 += (a == 0) ? d : 0.f;
            bb1 += (a == 1) ? d : 0.f;
            bb2 += (a == 2) ? d : 0.f;
            bb3 += (a == 3) ? d : 0.f;
            bb1 -= (a == 0) ? d : 0.f;
            bb2 -= (a == 1) ? d : 0.f;
            bb3 -= (a == 2) ? d : 0.f;
            bb4 -= (a == 3) ? d : 0.f;
        }
        float wk[VD];
        wk[0] = 1.f + bb0 + bb4; wk[1] = bb1; wk[2] = bb2; wk[3] = bb3;

        const int ri0 = (int)rintf(r0[0]);
        const int ri1 = (int)rintf(r0[1]);
        const int ri2 = (int)rintf(r0[2]);

        float f0 = 0.f, f1 = 0.f;
        const float* tbl = tables + (size_t)l * (size_t)CAP * 2;
#pragma unroll
        for (int k = 0; k < VD; ++k) {
            const int k0 = ri0 + k - ((rk[0] > NPOS - k) ? VD : 0);
            const int k1 = ri1 + k - ((rk[1] > NPOS - k) ? VD : 0);
            const int k2 = ri2 + k - ((rk[2] > NPOS - k) ? VD : 0);
            const unsigned h = (unsigned)k0
                             ^ ((unsigned)k1 * 2654435761u)
                             ^ ((unsigned)k2 * 805459861u);
            const unsigned idx = h & (unsigned)(CAP - 1);
            v2f g = *(const v2f*)(tbl + (size_t)idx * 2);   // 8B-aligned L2 gather
            f0 += g.x * wk[k];
            f1 += g.y * wk[k];
        }
        frow[2 * l]     = f0 * wl;
        frow[2 * l + 1] = f1 * wl;
    }
    frow[48] = px * 0.001f;
    frow[49] = py * 0.001f;
    frow[50] = pz * 0.001f;
#pragma unroll
    for (int c = 51; c < 64; ++c) frow[c] = 0.f;

    __syncthreads();

    // ---------------- Phase 2: MLP via V_WMMA_F32_16X16X4_F32 ------------------------
    const int lo = lane & 15;
    const int hi = lane >> 4;

    for (int tile = 0; tile < 2; ++tile) {
        const float (*X)[64] = (const float (*)[64])&feat[wv][tile * 16];

        // Layer 1: (16 x 52) @ (51 x 32), zero-padded K
        for (int nb = 0; nb < HID; nb += 16) {
            v8f acc = {0.f, 0.f, 0.f, 0.f, 0.f, 0.f, 0.f, 0.f};
            const int col = nb + lo;
#pragma unroll
            for (int k0 = 0; k0 < KPAD; k0 += 4) {
                const int kb = k0 + 2 * hi;
                v2f a = *(const v2f*)&X[lo][kb];
                v2f b;
                b.x = (kb     < INDIM) ? w1[kb * HID + col]       : 0.f;
                b.y = (kb + 1 < INDIM) ? w1[(kb + 1) * HID + col] : 0.f;
                acc = __builtin_amdgcn_wmma_f32_16x16x4_f32(false, a, false, b,
                                                            (short)0, acc, false, false);
            }
            const float bias = b1[col];
#pragma unroll
            for (int r = 0; r < 8; ++r)
                act[wv][0][8 * hi + r][col] = gelu_exact(acc[r] + bias);
        }
        __syncthreads();

        mid_layer(act[wv][0], w2, b2, act[wv][1], lo, hi);
        __syncthreads();
        mid_layer(act[wv][1], w3, b3, act[wv][0], lo, hi);
        __syncthreads();

        // Layer 4: (16 x 32) @ (32 x 33) -> write straight to global
        for (int nb = 0; nb < 48; nb += 16) {
            v8f acc = {0.f, 0.f, 0.f, 0.f, 0.f, 0.f, 0.f, 0.f};
            const int col = nb + lo;
            const bool okc = col < OUTD;
#pragma unroll
            for (int k0 = 0; k0 < HID; k0 += 4) {
                const int kb = k0 + 2 * hi;
                v2f a = *(const v2f*)&act[wv][0][lo][kb];
                v2f b;
                b.x = okc ? w4[kb * OUTD + col]       : 0.f;
                b.y = okc ? w4[(kb + 1) * OUTD + col] : 0.f;
                acc = __builtin_amdgcn_wmma_f32_16x16x4_f32(false, a, false, b,
                                                            (short)0, acc, false, false);
            }
            if (okc) {
                const float bias = b4[col];
#pragma unroll
                for (int r = 0; r < 8; ++r) {
                    const int row = tile * 16 + 8 * hi + r;
                    const int pt  = blockIdx.x * 128 + wv * 32 + row;
                    if (pt < N) {
                        const float v = acc[r] + bias;
                        if (col == 0) out[pt] = v;                                  // sdf
                        else out[(size_t)N + (size_t)pt * GEOM + (col - 1)] = v;    // geom
                    }
                }
            }
        }
        __syncthreads();
    }
}

extern "C" void kernel_launch(void* const* d_in, const int* in_sizes, int n_in,
                              void* d_out, int out_size, void* d_ws, size_t ws_size,
                              hipStream_t stream) {
    const float* points  = (const float*)d_in[0];
    const int*   iter_nr = (const int*)  d_in[1];
    const float* tables  = (const float*)d_in[2];
    const float* shifts  = (const float*)d_in[3];
    const float* w1 = (const float*)d_in[4];
    const float* b1 = (const float*)d_in[5];
    const float* w2 = (const float*)d_in[6];
    const float* b2 = (const float*)d_in[7];
    const float* w3 = (const float*)d_in[8];
    const float* b3 = (const float*)d_in[9];
    const float* w4 = (const float*)d_in[10];
    const float* b4 = (const float*)d_in[11];

    const int N = in_sizes[0] / 3;
    const int blocks = (N + 127) / 128;
    sdf_permuto_fused<<<blocks, 128, 0, stream>>>(points, iter_nr, tables, shifts,
                                                  w1, b1, w2, b2, w3, b3, w4, b4,
                                                  (float*)d_out, N);
}